// GNN_76158360092902
// MI455X (gfx1250) — compile-verified
//
#include <hip/hip_runtime.h>

// ---------------------------------------------------------------------------
// 2-layer GCN forward for MI455X (gfx1250, wave32).
//   h0 = x @ W1                       (FP32 WMMA 16x16x4, K=256)
//   agg1[d] += h0[s]*dinv[s]*dinv[d]  (hw f32 global atomics; self-loop folded
//                                      into init pass)
//   h1 = relu(agg1 + b1)              (folded into GEMM2 A-fragment load)
//   h2 = h1 @ W2                      (FP32 WMMA, K=16, N=40 in 3 tiles)
//   out[d] += h2[s]*norm  (+ b2, self-loop folded into init)
// Scatter phases dominate (~1.4 GB L2-resident traffic); GEMMs are <1 GFLOP,
// so fp32 WMMA (exact precision) beats fp16 WMMA (8x FLOP/instr we don't need).
// ---------------------------------------------------------------------------

#define N_NODES  100000
#define N_EDGES  3200000
#define F_IN     256
#define HID      16
#define NCLS     40
#define N_TILES  (N_NODES / 16)   // 6250 exactly

typedef float v2f __attribute__((ext_vector_type(2)));
typedef float v8f __attribute__((ext_vector_type(8)));

// ---------------- degree / normalization ----------------------------------

__global__ void k_deg_init(float* deg) {
    int i = blockIdx.x * blockDim.x + threadIdx.x;
    if (i < N_NODES) deg[i] = 1.0f;          // self-loop contribution
}

__global__ void k_deg_edges(const long long* __restrict__ dst, float* deg) {
    int e = blockIdx.x * blockDim.x + threadIdx.x;
    if (e < N_EDGES) unsafeAtomicAdd(&deg[(int)dst[e]], 1.0f);
}

__global__ void k_dinv(float* deg_dinv) {
    int i = blockIdx.x * blockDim.x + threadIdx.x;
    if (i < N_NODES) {
        float d = deg_dinv[i];
        deg_dinv[i] = (d > 0.0f) ? rsqrtf(d) : 0.0f;
    }
}

// ---------------- GEMM1: h0 = x @ W1  (WMMA f32 16x16x4) -------------------
// One wave per 16-row tile; 4 waves per block. W1 (16 KB) staged in LDS.

__global__ __launch_bounds__(128) void k_gemm1(const float* __restrict__ x,
                                               const float* __restrict__ W1,
                                               float* __restrict__ h0) {
    __shared__ float sW[F_IN * HID];         // 16 KB of the 320 KB WGP LDS
    for (int i = threadIdx.x; i < F_IN * HID; i += 128) sW[i] = W1[i];
    __syncthreads();

    const int wave = threadIdx.x >> 5;
    const int lane = threadIdx.x & 31;
    const int tile = blockIdx.x * 4 + wave;
    if (tile >= N_TILES) return;             // wave-uniform exit (EXEC stays full)

    const int base  = tile * 16;
    const int m     = lane & 15;             // row (A) / col (B,C)
    const int khalf = (lane >> 4) ? 2 : 0;   // lane half selects K pair
    const float* __restrict__ arow = x + (size_t)(base + m) * F_IN;

    v8f c = {0.f,0.f,0.f,0.f,0.f,0.f,0.f,0.f};
    #pragma unroll 8
    for (int k = 0; k < F_IN; k += 4) {
        v2f a, b;
        a.x = arow[k + khalf];               // A: 16x4 f32, K striped by lane half
        a.y = arow[k + khalf + 1];
        b.x = sW[(k + khalf)     * HID + m]; // B: 4x16 f32, same K striping
        b.y = sW[(k + khalf + 1) * HID + m];
        c = __builtin_amdgcn_wmma_f32_16x16x4_f32(false, a, false, b,
                                                  (short)0, c, false, false);
    }

    const int mofs = (lane >> 4) ? 8 : 0;    // C/D: VGPR r -> rows r / r+8
    #pragma unroll
    for (int r = 0; r < 8; ++r)
        h0[(size_t)(base + r + mofs) * HID + m] = c[r];
}

// ---------------- layer-1 aggregation --------------------------------------

__global__ void k_agg1_init(const float* __restrict__ h0,
                            const float* __restrict__ dinv,
                            float* __restrict__ agg1) {
    int idx = blockIdx.x * blockDim.x + threadIdx.x;   // node*16 + f
    if (idx < N_NODES * HID) {
        int i = idx >> 4;
        float di = dinv[i];
        agg1[idx] = h0[idx] * di * di;       // self-loop term, no atomic needed
    }
}

__global__ void k_agg1_edges(const long long* __restrict__ src,
                             const long long* __restrict__ dst,
                             const float* __restrict__ h0,
                             const float* __restrict__ dinv,
                             float* __restrict__ agg1) {
    int idx = blockIdx.x * blockDim.x + threadIdx.x;   // edge*16 + f
    if (idx < N_EDGES * HID) {
        int e = idx >> 4, f = idx & 15;
        int s = (int)src[e], d = (int)dst[e];
        float norm = dinv[s] * dinv[d];
        unsafeAtomicAdd(&agg1[(size_t)d * HID + f],
                        h0[(size_t)s * HID + f] * norm);
    }
}

// ---------------- GEMM2: h2 = relu(agg1+b1) @ W2  (WMMA f32) ---------------
// K=16 (4 wmma steps), N=40 handled as 3 column tiles (last one masked).

__global__ __launch_bounds__(128) void k_gemm2(const float* __restrict__ agg1,
                                               const float* __restrict__ b1,
                                               const float* __restrict__ W2,
                                               float* __restrict__ h2) {
    __shared__ float sW[HID * NCLS];         // 2.5 KB
    __shared__ float sB1[HID];
    for (int i = threadIdx.x; i < HID * NCLS; i += 128) sW[i] = W2[i];
    if (threadIdx.x < HID) sB1[threadIdx.x] = b1[threadIdx.x];
    __syncthreads();

    const int wave = threadIdx.x >> 5;
    const int lane = threadIdx.x & 31;
    const int tile = blockIdx.x * 4 + wave;
    if (tile >= N_TILES) return;

    const int base  = tile * 16;
    const int m     = lane & 15;
    const int khalf = (lane >> 4) ? 2 : 0;
    const int mofs  = (lane >> 4) ? 8 : 0;

    // A fragments for all 4 K-steps, with relu(.+b1) fused at load.
    v2f afr[4];
    #pragma unroll
    for (int ks = 0; ks < 4; ++ks) {
        int kk = ks * 4 + khalf;
        float a0 = agg1[(size_t)(base + m) * HID + kk]     + sB1[kk];
        float a1 = agg1[(size_t)(base + m) * HID + kk + 1] + sB1[kk + 1];
        afr[ks].x = fmaxf(a0, 0.0f);
        afr[ks].y = fmaxf(a1, 0.0f);
    }

    #pragma unroll
    for (int nt = 0; nt < 3; ++nt) {
        const int  n  = nt * 16 + m;
        const bool nv = (n < NCLS);
        v8f c = {0.f,0.f,0.f,0.f,0.f,0.f,0.f,0.f};
        #pragma unroll
        for (int ks = 0; ks < 4; ++ks) {
            int kk = ks * 4 + khalf;
            v2f b;
            b.x = nv ? sW[kk       * NCLS + n] : 0.0f;   // lane-masked load only;
            b.y = nv ? sW[(kk + 1) * NCLS + n] : 0.0f;   // wmma runs at full EXEC
            c = __builtin_amdgcn_wmma_f32_16x16x4_f32(false, afr[ks], false, b,
                                                      (short)0, c, false, false);
        }
        if (nv) {
            #pragma unroll
            for (int r = 0; r < 8; ++r)
                h2[(size_t)(base + r + mofs) * NCLS + n] = c[r];
        }
    }
}

// ---------------- layer-2 aggregation + bias -------------------------------

__global__ void k_out_init(const float* __restrict__ h2,
                           const float* __restrict__ dinv,
                           const float* __restrict__ b2,
                           float* __restrict__ out) {
    int idx = blockIdx.x * blockDim.x + threadIdx.x;   // node*40 + f
    if (idx < N_NODES * NCLS) {
        int i = idx / NCLS, f = idx - i * NCLS;
        float di = dinv[i];
        out[idx] = h2[idx] * di * di + b2[f];          // self-loop + bias
    }
}

__global__ void k_out_edges(const long long* __restrict__ src,
                            const long long* __restrict__ dst,
                            const float* __restrict__ h2,
                            const float* __restrict__ dinv,
                            float* __restrict__ out) {
    int idx = blockIdx.x * blockDim.x + threadIdx.x;   // edge*40 + f
    if (idx < N_EDGES * NCLS) {
        int e = idx / NCLS, f = idx - e * NCLS;
        int s = (int)src[e], d = (int)dst[e];
        float norm = dinv[s] * dinv[d];
        unsafeAtomicAdd(&out[(size_t)d * NCLS + f],
                        h2[(size_t)s * NCLS + f] * norm);
    }
}

// ---------------------------------------------------------------------------

extern "C" void kernel_launch(void* const* d_in, const int* in_sizes, int n_in,
                              void* d_out, int out_size, void* d_ws, size_t ws_size,
                              hipStream_t stream) {
    const float*     x    = (const float*)d_in[0];
    const long long* ei   = (const long long*)d_in[1];   // int64 per reference
    const float*     W1   = (const float*)d_in[2];
    const float*     b1   = (const float*)d_in[3];
    const float*     W2   = (const float*)d_in[4];
    const float*     b2   = (const float*)d_in[5];
    float*           out  = (float*)d_out;

    const long long* esrc = ei;              // edge_index[0]
    const long long* edst = ei + N_EDGES;    // edge_index[1]

    // workspace layout (floats): dinv | h0 | agg1 | h2   (~30 MB total)
    float* ws   = (float*)d_ws;
    float* dinv = ws;                          // 100000
    float* h0   = ws + 131072;                 // 1,600,000
    float* agg1 = h0 + (size_t)N_NODES * HID;  // 1,600,000
    float* h2   = agg1 + (size_t)N_NODES * HID;// 4,000,000

    const int T = 256;

    k_deg_init  <<<(N_NODES + T - 1) / T, T, 0, stream>>>(dinv);
    k_deg_edges <<<(N_EDGES + T - 1) / T, T, 0, stream>>>(edst, dinv);
    k_dinv      <<<(N_NODES + T - 1) / T, T, 0, stream>>>(dinv);

    k_gemm1     <<<(N_TILES + 3) / 4, 128, 0, stream>>>(x, W1, h0);

    k_agg1_init <<<(N_NODES * HID + T - 1) / T, T, 0, stream>>>(h0, dinv, agg1);
    k_agg1_edges<<<(N_EDGES * HID + T - 1) / T, T, 0, stream>>>(esrc, edst, h0, dinv, agg1);

    k_gemm2     <<<(N_TILES + 3) / 4, 128, 0, stream>>>(agg1, b1, W2, h2);

    k_out_init  <<<(N_NODES * NCLS + T - 1) / T, T, 0, stream>>>(h2, dinv, b2, out);
    k_out_edges <<<(N_EDGES * NCLS + T - 1) / T, T, 0, stream>>>(esrc, edst, h2, dinv, out);
}